// PushModel_40046275068144
// MI455X (gfx1250) — compile-verified
//
#include <hip/hip_runtime.h>
#include <hip/hip_bf16.h>
#include <math.h>

#define VOCAB 14
#define TOKD 8
#define POSD 8
#define DM 16
#define NH 2
#define HD 3
#define FFD 3
#define BB 16
#define TT 2048
#define EPSF 1e-5f

typedef __attribute__((ext_vector_type(2))) float v2f;
typedef __attribute__((ext_vector_type(8))) float v8f;

__device__ __forceinline__ void layernorm16(const float* xin, float* out,
                                            const float* w, const float* b) {
  float m = 0.f;
#pragma unroll
  for (int j = 0; j < 16; ++j) m += xin[j];
  m *= (1.f / 16.f);
  float v = 0.f;
#pragma unroll
  for (int j = 0; j < 16; ++j) { float d = xin[j] - m; v += d * d; }
  v *= (1.f / 16.f);
  float r = rsqrtf(v + EPSF);
#pragma unroll
  for (int j = 0; j < 16; ++j) out[j] = (xin[j] - m) * r * w[j] + b[j];
}

// ---------------- Kernel 1: embedding + LN1 + QKV projections ----------------
// q,k stored (B*H, T, 4) with dim3 = 0 pad; v stored (B*H, T, 4) with dim3 = 1.0
// (the ones column makes the PV WMMA also accumulate the softmax denominator).
__global__ void __launch_bounds__(256) k_embed_qkv(
    const int* __restrict__ idx, const float* __restrict__ tok_emb,
    const float* __restrict__ pos_enc, const float* __restrict__ Wq,
    const float* __restrict__ Wk, const float* __restrict__ Wv,
    const float* __restrict__ ln1w, const float* __restrict__ ln1b,
    float* __restrict__ xres, float* __restrict__ qb, float* __restrict__ kb,
    float* __restrict__ vb) {
  int tid = blockIdx.x * blockDim.x + threadIdx.x;
  if (tid >= BB * TT) return;
  int t = tid & (TT - 1);
  int b = tid >> 11;
  int tok = idx[tid];
  float x[16], h[16];
#pragma unroll
  for (int j = 0; j < 8; ++j) x[j] = tok_emb[tok * 8 + j];
#pragma unroll
  for (int j = 0; j < 8; ++j) x[8 + j] = pos_enc[t * 8 + j];
#pragma unroll
  for (int j = 0; j < 16; ++j) xres[tid * 16 + j] = x[j];
  layernorm16(x, h, ln1w, ln1b);
#pragma unroll
  for (int hh = 0; hh < NH; ++hh) {
    int bh = b * NH + hh;
    int base = (bh * TT + t) * 4;
#pragma unroll
    for (int i = 0; i < 4; ++i) {
      float qa = 0.f, ka = 0.f, va = 0.f;
      if (i < 3) {
#pragma unroll
        for (int j = 0; j < 8; ++j) {
          qa += Wq[(hh * 3 + i) * 8 + j] * h[8 + j];
          ka += Wk[(hh * 3 + i) * 8 + j] * h[8 + j];
          va += Wv[(hh * 3 + i) * 8 + j] * h[j];
        }
      } else {
        va = 1.0f;  // ones column -> softmax denominator via WMMA
      }
      qb[base + i] = qa;
      kb[base + i] = ka;
      vb[base + i] = va;
    }
  }
}

// ---------------- Kernel 2: causal attention, WMMA f32 16x16x4 ----------------
// One wave32 per (bh, 16-query tile). Sᵀ = K·Qᵀ, P = exp2(mask(Sᵀ·log2e/√3)),
// outᵀ = Vᵀ·Pᵀ accumulated in 4 K=4 chunks; row 3 of outᵀ = denominators.
// Interior key tiles are mask-free; the diagonal tile is peeled (MASKED=true).
// V A-operand loads are UNPREDICATED: lanes with d=n>=4 feed only outᵀ rows
// 4..15, which are never read (we consume acc[0..3] on lanes 0..15 = rows
// 0..3). WMMA computes each D element independently per A-row, so junk rows
// are harmless; the tiny address overrun past vb lands in the adjacent ao
// region of the workspace (mapped, unused values).

template <bool MASKED>
__device__ __forceinline__ void attn_tile(int jt, int vbase, int n, int hi,
                                          int qg, v2f aq, const v2f* k2,
                                          const float* vb, v8f& acc) {
  const float lscale = 0.8329404163f;  // log2(e) / sqrt(3)
  int kt16 = vbase + jt * 16;
  v2f bk = k2[(kt16 + n) * 2 + hi];  // K tile as A-operand (M=key)

  v8f c0 = {};
  // Sᵀ[key][query], key = jt*16 + r + 8*hi, query = qt*16 + n
  v8f s = __builtin_amdgcn_wmma_f32_16x16x4_f32(false, bk, false, aq, (short)0,
                                                c0, false, false);
  float p[8], sh[8];
#pragma unroll
  for (int r = 0; r < 8; ++r) {
    // raw hw exp2: one v_mul + one v_exp_f32, no libm range guard
    float e = __builtin_amdgcn_exp2f(s[r] * lscale);  // == exp(s/sqrt(3))
    if (MASKED) {
      int kg = jt * 16 + r + 8 * hi;
      p[r] = (kg <= qg) ? e : 0.0f;  // causal mask (diagonal tile only)
    } else {
      p[r] = e;
    }
  }
#pragma unroll
  for (int r = 0; r < 8; ++r) sh[r] = __shfl_xor(p[r], 16, 32);

  // Pᵀ chunks as B-operands (K = 4 keys x N = 16 queries), keys 4c + r + 2*hi
  v2f bp0 = {hi ? sh[2] : p[0], hi ? sh[3] : p[1]};  // keys 0..3
  v2f bp1 = {hi ? sh[6] : p[4], hi ? sh[7] : p[5]};  // keys 4..7
  v2f bp2 = {hi ? p[2] : sh[0], hi ? p[3] : sh[1]};  // keys 8..11
  v2f bp3 = {hi ? p[6] : sh[4], hi ? p[7] : sh[5]};  // keys 12..15

  // Vᵀ chunks as A-operands: M = dim = n; K = 4c + {0,1} + 2*hi.
  // Unpredicated loads (see note above).
  int d = n;
#define VLOAD(koff) (vb[(kt16 + (koff)) * 4 + d])
  v2f av;
  av.x = VLOAD(0 + 2 * hi);  av.y = VLOAD(1 + 2 * hi);
  acc = __builtin_amdgcn_wmma_f32_16x16x4_f32(false, av, false, bp0, (short)0,
                                              acc, false, false);
  av.x = VLOAD(4 + 2 * hi);  av.y = VLOAD(5 + 2 * hi);
  acc = __builtin_amdgcn_wmma_f32_16x16x4_f32(false, av, false, bp1, (short)0,
                                              acc, false, false);
  av.x = VLOAD(8 + 2 * hi);  av.y = VLOAD(9 + 2 * hi);
  acc = __builtin_amdgcn_wmma_f32_16x16x4_f32(false, av, false, bp2, (short)0,
                                              acc, false, false);
  av.x = VLOAD(12 + 2 * hi); av.y = VLOAD(13 + 2 * hi);
  acc = __builtin_amdgcn_wmma_f32_16x16x4_f32(false, av, false, bp3, (short)0,
                                              acc, false, false);
#undef VLOAD
}

__global__ void __launch_bounds__(128) k_attn(const float* __restrict__ qb,
                                              const float* __restrict__ kb,
                                              const float* __restrict__ vb,
                                              float* __restrict__ ao) {
  int wave = blockIdx.x * 4 + (threadIdx.x >> 5);
  // Force wave-uniform values into SGPRs: scalar loop, EXEC all-1s for WMMA.
  int bh = __builtin_amdgcn_readfirstlane(wave >> 7);   // / (T/16)
  int qt = __builtin_amdgcn_readfirstlane(wave & 127);  // query tile index
  int lane = threadIdx.x & 31;
  int n = lane & 15;
  int hi = lane >> 4;  // lane half: selects K pair {0,1} vs {2,3}

  const v2f* q2 = (const v2f*)qb;
  const v2f* k2 = (const v2f*)kb;
  int vbase = bh * TT;

  // Q as B-operand of Sᵀ = K·Qᵀ: lane n = query column, vgpr pair = dims.
  v2f aq = q2[(vbase + qt * 16 + n) * 2 + hi];

  v8f acc = {};
  int qg = qt * 16 + n;

  // Interior tiles: every key <= every query in the tile -> no mask.
  for (int jt = 0; jt < qt; ++jt) {
    int nxt = (vbase + (jt + 1) * 16 + n) * 4;
    __builtin_prefetch(&kb[nxt], 0, 3);  // global_prefetch_b8 (next K tile)
    __builtin_prefetch(&vb[nxt], 0, 3);  // global_prefetch_b8 (next V tile)
    attn_tile<false>(jt, vbase, n, hi, qg, aq, k2, vb, acc);
  }
  // Diagonal tile: causal mask applies.
  attn_tile<true>(qt, vbase, n, hi, qg, aq, k2, vb, acc);

  // outᵀ rows 0..2 = unnormalized out dims, row 3 = denominator; lanes 0-15
  if (lane < 16) {
    float inv = 1.0f / acc[3];
    int orow = (vbase + qt * 16 + lane) * 4;
    ao[orow + 0] = acc[0] * inv;
    ao[orow + 1] = acc[1] * inv;
    ao[orow + 2] = acc[2] * inv;
  }
}

// ---------------- Kernel 3: Wo proj + LN2 + FFN(GELU) + LNf + vocab head ----
__global__ void __launch_bounds__(256) k_tail(
    const float* __restrict__ xres, const float* __restrict__ ao,
    const float* __restrict__ Wo, const float* __restrict__ ln2w,
    const float* __restrict__ ln2b, const float* __restrict__ lnfw,
    const float* __restrict__ lnfb, const float* __restrict__ W1,
    const float* __restrict__ b1, const float* __restrict__ W2,
    const float* __restrict__ b2, const float* __restrict__ Wh,
    const float* __restrict__ tok_emb, float* __restrict__ out) {
  int tid = blockIdx.x * blockDim.x + threadIdx.x;
  if (tid >= BB * TT) return;
  int t = tid & (TT - 1);
  int b = tid >> 11;
  float x[16];
#pragma unroll
  for (int j = 0; j < 16; ++j) x[j] = xres[tid * 16 + j];
  float o6[6];
#pragma unroll
  for (int hh = 0; hh < NH; ++hh)
#pragma unroll
    for (int i = 0; i < 3; ++i)
      o6[hh * 3 + i] = ao[((b * NH + hh) * TT + t) * 4 + i];
#pragma unroll
  for (int j = 0; j < 16; ++j) {
    float a = 0.f;
#pragma unroll
    for (int i = 0; i < 6; ++i) a += Wo[j * 6 + i] * o6[i];
    x[j] += a;
  }
  float h2[16];
  layernorm16(x, h2, ln2w, ln2b);
  float g[3];
#pragma unroll
  for (int i = 0; i < 3; ++i) {
    float f = b1[i];
#pragma unroll
    for (int j = 0; j < 16; ++j) f += W1[i * 16 + j] * h2[j];
    g[i] = 0.5f * f * (1.0f + erff(f * 0.70710678118f));  // exact GELU
  }
#pragma unroll
  for (int j = 0; j < 16; ++j) {
    float a = b2[j];
#pragma unroll
    for (int i = 0; i < 3; ++i) a += W2[j * 3 + i] * g[i];
    x[j] += a;
  }
  float xf[16];
  layernorm16(x, xf, lnfw, lnfb);
  float y[8];
#pragma unroll
  for (int d = 0; d < 8; ++d) {
    float a = 0.f;
#pragma unroll
    for (int j = 0; j < 16; ++j) a += Wh[d * 16 + j] * xf[j];
    y[d] = a;
  }
#pragma unroll
  for (int vv = 0; vv < VOCAB; ++vv) {
    float a = 0.f;
#pragma unroll
    for (int d = 0; d < 8; ++d) a += tok_emb[vv * 8 + d] * y[d];
    out[tid * VOCAB + vv] = a;
  }
}

extern "C" void kernel_launch(void* const* d_in, const int* in_sizes, int n_in,
                              void* d_out, int out_size, void* d_ws,
                              size_t ws_size, hipStream_t stream) {
  const int* idx = (const int*)d_in[0];
  const float* tok_emb = (const float*)d_in[1];
  const float* pos_enc = (const float*)d_in[2];
  const float* Wq = (const float*)d_in[3];
  const float* Wk = (const float*)d_in[4];
  const float* Wv = (const float*)d_in[5];
  const float* Wo = (const float*)d_in[6];
  const float* ln1w = (const float*)d_in[7];
  const float* ln1b = (const float*)d_in[8];
  const float* ln2w = (const float*)d_in[9];
  const float* ln2b = (const float*)d_in[10];
  const float* lnfw = (const float*)d_in[11];
  const float* lnfb = (const float*)d_in[12];
  const float* W1 = (const float*)d_in[13];
  const float* b1 = (const float*)d_in[14];
  const float* W2 = (const float*)d_in[15];
  const float* b2 = (const float*)d_in[16];
  const float* Wh = (const float*)d_in[17];
  float* out = (float*)d_out;

  float* ws = (float*)d_ws;
  size_t ntok = (size_t)BB * TT;           // 32768
  size_t qkvN = (size_t)BB * NH * TT * 4;  // 262144 floats each
  float* xres = ws;                        // 524288 floats
  float* qb = xres + ntok * 16;
  float* kb = qb + qkvN;
  float* vb = kb + qkvN;
  float* ao = vb + qkvN;  // total 6 MB of the workspace

  k_embed_qkv<<<(int)(ntok / 256), 256, 0, stream>>>(
      idx, tok_emb, pos_enc, Wq, Wk, Wv, ln1w, ln1b, xres, qb, kb, vb);

  int tiles = BB * NH * (TT / 16);  // 4096 waves, 4 per 128-thread block
  k_attn<<<tiles / 4, 128, 0, stream>>>(qb, kb, vb, ao);

  k_tail<<<(int)(ntok / 256), 256, 0, stream>>>(xres, ao, Wo, ln2w, ln2b, lnfw,
                                                lnfb, W1, b1, W2, b2, Wh,
                                                tok_emb, out);
}